// SpecialSpmmFinal_84859963834577
// MI455X (gfx1250) — compile-verified
//
#include <hip/hip_runtime.h>
#include <stdint.h>

// ---------------------------------------------------------------------------
// segment_sum(edge_w[E,64], edge[0][E]) -> out[N,64]   (scatter-add, f32)
//
// Memory-bound: 320 MB streamed reads + 80M f32 atomics into a 12.8 MB
// output that lives in L2. Strategy:
//   * async-copy edge_w tiles global->LDS (ASYNCcnt, double buffered)
//   * one wave32 per edge row: 32 lanes x float2 = 256 B row
//   * global_atomic_add_f32 (no-return, agent scope) into out
// ---------------------------------------------------------------------------

#define OUT_F            64
#define WAVES_PER_BLOCK  8
#define BLOCK_THREADS    (WAVES_PER_BLOCK * 32)       // 256
#define EDGES_PER_WAVE   8
#define TILE_EDGES       (WAVES_PER_BLOCK * EDGES_PER_WAVE)  // 64 edges/tile
#define ROW_BYTES        (OUT_F * 4)                  // 256 B per edge row
#define SLICE_BYTES      (EDGES_PER_WAVE * ROW_BYTES) // 2 KB per wave slice
#define TILE_BYTES       (TILE_EDGES * ROW_BYTES)     // 16 KB per tile

// Issue 4 async b128 loads: this lane's 64 bytes of the wave's 2 KB slice.
// INST_OFFSET is added to BOTH the LDS and global addresses (ISA 08 §4.4),
// so one (lds, gaddr) register pair covers all four 16 B chunks.
__device__ __forceinline__ void issue_async_slice(const float* __restrict__ edge_w,
                                                  uint32_t tile, uint32_t wave,
                                                  uint32_t lane, uint32_t lds_base) {
    uint64_t g = (uint64_t)edge_w + (uint64_t)tile * TILE_BYTES
               + (uint64_t)wave * SLICE_BYTES + (uint64_t)lane * 64u;
    uint32_t l = lds_base + wave * SLICE_BYTES + lane * 64u;
    asm volatile(
        "global_load_async_to_lds_b128 %0, %1, off\n\t"
        "global_load_async_to_lds_b128 %0, %1, off offset:16\n\t"
        "global_load_async_to_lds_b128 %0, %1, off offset:32\n\t"
        "global_load_async_to_lds_b128 %0, %1, off offset:48"
        :: "v"(l), "v"(g) : "memory");
}

__device__ __forceinline__ void atomic_add2(float* p, float2 v) {
    __hip_atomic_fetch_add(p + 0, v.x, __ATOMIC_RELAXED, __HIP_MEMORY_SCOPE_AGENT);
    __hip_atomic_fetch_add(p + 1, v.y, __ATOMIC_RELAXED, __HIP_MEMORY_SCOPE_AGENT);
}

__global__ void __launch_bounds__(BLOCK_THREADS)
segment_sum_scatter_kernel(const int* __restrict__ edge_src,   // edge[0], E ints
                           const float* __restrict__ edge_w,   // [E, 64]
                           float* __restrict__ out,            // [N, 64]
                           uint32_t E) {
    __shared__ __align__(16) unsigned char lds_buf[2 * TILE_BYTES];  // 32 KB

    const uint32_t lane = threadIdx.x & 31u;
    const uint32_t wave = threadIdx.x >> 5;
    const uint32_t num_full_tiles = E / TILE_EDGES;
    const uint32_t stride = gridDim.x;

    // Generic LDS pointer = {shared_aperture_hi32, lds_offset}; low 32 bits
    // are the wave-relative LDS byte offset the async instruction needs.
    const uint32_t lds_base = (uint32_t)(uintptr_t)(void*)lds_buf;

    uint32_t t = blockIdx.x;
    uint32_t buf = 0;
    if (t < num_full_tiles) {
        issue_async_slice(edge_w, t, wave, lane, lds_base);          // prologue
    }
    for (; t < num_full_tiles; t += stride) {
        const uint32_t tn = t + stride;
        if (tn < num_full_tiles) {
            issue_async_slice(edge_w, tn, wave, lane, lds_base + (buf ^ 1u) * TILE_BYTES);
            // 4 in flight for next tile; <=4 means current tile's 4 are done
            asm volatile("s_wait_asynccnt 0x4" ::: "memory");
        } else {
            asm volatile("s_wait_asynccnt 0x0" ::: "memory");
        }

        // Consume this wave's slice: 8 edges, lane holds float2 of each row.
        const float2* rows = (const float2*)(lds_buf + buf * TILE_BYTES + wave * SLICE_BYTES);
        const uint32_t ebase = t * TILE_EDGES + wave * EDGES_PER_WAVE;
        #pragma unroll
        for (int j = 0; j < EDGES_PER_WAVE; ++j) {
            const int idx = edge_src[ebase + j];            // wave-uniform
            const float2 v = rows[j * 32 + lane];           // ds_load_b64, conflict-free
            atomic_add2(out + (size_t)idx * OUT_F + lane * 2u, v);
        }
        buf ^= 1u;
    }

    // Tail (< TILE_EDGES edges): direct global load -> atomic path.
    const uint32_t gwave = blockIdx.x * WAVES_PER_BLOCK + wave;
    const uint32_t nwave = gridDim.x * WAVES_PER_BLOCK;
    for (uint32_t e = num_full_tiles * TILE_EDGES + gwave; e < E; e += nwave) {
        const int idx = edge_src[e];
        const float2 v = *(const float2*)(edge_w + (size_t)e * OUT_F + lane * 2u);
        atomic_add2(out + (size_t)idx * OUT_F + lane * 2u, v);
    }
}

__global__ void zero_fill_f32x4_kernel(float4* __restrict__ out, uint32_t n4) {
    uint32_t i = blockIdx.x * blockDim.x + threadIdx.x;
    const uint32_t s = gridDim.x * blockDim.x;
    const float4 z = make_float4(0.f, 0.f, 0.f, 0.f);
    for (; i < n4; i += s) out[i] = z;
}

extern "C" void kernel_launch(void* const* d_in, const int* in_sizes, int n_in,
                              void* d_out, int out_size, void* d_ws, size_t ws_size,
                              hipStream_t stream) {
    (void)n_in; (void)d_ws; (void)ws_size;
    const int*   edge   = (const int*)d_in[0];     // [2, E]; row 0 = source idx
    const float* edge_w = (const float*)d_in[1];   // [E, 64]
    const uint32_t E = (uint32_t)(in_sizes[0] / 2);
    float* out = (float*)d_out;                    // [N, 64]

    // Harness poisons d_out: zero it first (out_size divisible by 4).
    const uint32_t n4 = (uint32_t)(out_size / 4);
    zero_fill_f32x4_kernel<<<512, 256, 0, stream>>>((float4*)out, n4);

    const uint32_t num_full_tiles = E / TILE_EDGES;
    uint32_t grid = num_full_tiles < 2048u ? (num_full_tiles ? num_full_tiles : 1u) : 2048u;
    segment_sum_scatter_kernel<<<grid, BLOCK_THREADS, 0, stream>>>(edge, edge_w, out, E);
}